// ProbAttention_34385508172159
// MI455X (gfx1250) — compile-verified
//
#include <hip/hip_runtime.h>

// Problem constants (from reference setup_inputs)
#define B_    4
#define L_    2048
#define H_    8
#define D_    64
#define NTOP  40     // u = U_part = 5*ceil(ln(2048)) = 40
#define MPAD  48     // 40 padded to 3 WMMA tiles of 16
#define NSAMP 40
#define KPADS 68     // LDS row stride (floats): 16B-aligned rows, 68%64=4 -> no bank conflicts

typedef __attribute__((ext_vector_type(2))) float v2f;
typedef __attribute__((ext_vector_type(8))) float v8f;

// -------------------------------------------------- CDNA5 async global->LDS
#if defined(__has_builtin)
#if __has_builtin(__builtin_amdgcn_global_load_async_to_lds_b128) && \
    __has_builtin(__builtin_amdgcn_s_wait_asynccnt)
#define HAVE_ASYNC 1
#endif
#endif
#ifndef HAVE_ASYNC
#define HAVE_ASYNC 0
#endif

typedef int v4i_ __attribute__((vector_size(16)));
typedef __attribute__((address_space(1))) v4i_* gv4p;   // global 16B-vector ptr
typedef __attribute__((address_space(3))) v4i_* lv4p;   // LDS    16B-vector ptr

__device__ __forceinline__ void async_copy_b128(const float* g, float* l) {
#if HAVE_ASYNC
  __builtin_amdgcn_global_load_async_to_lds_b128((gv4p)(float*)g, (lv4p)l, 0, 0);
#else
  *(float4*)l = *(const float4*)g;
#endif
}

__device__ __forceinline__ void async_wait() {
#if HAVE_ASYNC
  __builtin_amdgcn_s_wait_asynccnt(0);
#endif
}

__device__ __forceinline__ v8f wmma4(v2f a, v2f b, v8f c) {
  return __builtin_amdgcn_wmma_f32_16x16x4_f32(false, a, false, b, (short)0,
                                               c, false, false);
}

// ---------------------------------------------------------------- utilities
__device__ __forceinline__ unsigned mix32(unsigned x) {
  x ^= x >> 16; x *= 0x7feb352dU;
  x ^= x >> 15; x *= 0x846ca68bU;
  x ^= x >> 16; return x;
}

// ---------------------------------------------------- 1) V mean per (b,h,d)
__global__ void k_vmean(const float* __restrict__ v, float* __restrict__ vmean) {
  int bh = blockIdx.x, b = bh >> 3, h = bh & 7;
  int d = threadIdx.x;                       // 64 threads
  const float* p = v + ((size_t)b * L_ * H_ + h) * D_ + d;
  float s = 0.f;
  for (int l = 0; l < L_; ++l) s += p[(size_t)l * H_ * D_];
  vmean[bh * D_ + d] = s * (1.0f / L_);
}

// --------------------------------- 2) broadcast V mean baseline (b128 wide)
__global__ void k_broadcast(float4* __restrict__ out4, const float4* __restrict__ vmean4) {
  unsigned i = blockIdx.x * 256u + threadIdx.x;  // total (B*L*H*D)/4 = 2^20
  unsigned c16 = i & 15;                         // 16 float4 per head row
  unsigned h = (i >> 4) & 7;
  unsigned b = i >> 18;
  out4[i] = vmean4[((b << 3) + h) * 16 + c16];
}

// --------------------------- 3) sampled Q*K dot products -> sparsity metric M
__global__ void k_sample_m(const float* __restrict__ q, const float* __restrict__ k,
                           float* __restrict__ M) {
  int bh = blockIdx.x, b = bh >> 3, h = bh & 7;
  int wave = threadIdx.x >> 5, lane = threadIdx.x & 31;
  int qi = blockIdx.y * 8 + wave;            // one wave32 per query row
  const float* qrow = q + (((size_t)b * L_ + qi) * H_ + h) * D_;
  float q0 = qrow[lane], q1 = qrow[lane + 32];
  const float* kb = k + ((size_t)b * L_ * H_ + h) * D_;
  float mx = -1e30f, sum = 0.f;
  for (int s = 0; s < NSAMP; ++s) {
    unsigned idx = mix32((unsigned)(qi * NSAMP + s) + 0x2545F491u) & (L_ - 1);
    const float* krow = kb + (size_t)idx * H_ * D_;
    float p = q0 * krow[lane] + q1 * krow[lane + 32];
    for (int o = 16; o > 0; o >>= 1) p += __shfl_xor(p, o, 32);
    mx = fmaxf(mx, p); sum += p;
  }
  if (lane == 0) M[bh * L_ + qi] = mx - sum * (1.0f / NSAMP);
}

// ------------------------------------------------- 4) top-40 of M per (b,h)
__global__ void k_topk(const float* __restrict__ M, int* __restrict__ mtop) {
  __shared__ float m[L_];
  __shared__ float rv[256];
  __shared__ int   ri[256];
  int bh = blockIdx.x, t = threadIdx.x;
  for (int i = t; i < L_; i += 256) m[i] = M[bh * L_ + i];
  __syncthreads();
  for (int it = 0; it < NTOP; ++it) {
    float bv = -1e30f; int bi = 0;
    for (int i = t; i < L_; i += 256) { float x = m[i]; if (x > bv) { bv = x; bi = i; } }
    rv[t] = bv; ri[t] = bi;
    __syncthreads();
    for (int s = 128; s > 0; s >>= 1) {
      if (t < s && rv[t + s] > rv[t]) { rv[t] = rv[t + s]; ri[t] = ri[t + s]; }
      __syncthreads();
    }
    if (t == 0) { mtop[bh * MPAD + it] = ri[0]; m[ri[0]] = -1e30f; }
    __syncthreads();
  }
}

// ------------------------------------------ 5) gather Q_reduce (zero padded)
__global__ void k_qred(const float* __restrict__ q, const int* __restrict__ mtop,
                       float* __restrict__ qred) {
  int bh = blockIdx.x, b = bh >> 3, h = bh & 7;
  for (int e = threadIdx.x; e < MPAD * D_; e += 256) {
    int row = e >> 6, d = e & 63;
    float val = 0.f;
    if (row < NTOP) {
      int l = mtop[bh * MPAD + row];
      val = q[(((size_t)b * L_ + l) * H_ + h) * D_ + d];
    }
    qred[(size_t)bh * MPAD * D_ + e] = val;
  }
}

// --------------------- 6) scores = Qred(48x64) * K^T, WMMA f32 16x16x4 tiles
__global__ void k_scores(const float* __restrict__ kmat, const float* __restrict__ qred,
                         float* __restrict__ scores) {
  __shared__ __align__(16) float Qs[MPAD][KPADS];
  __shared__ __align__(16) float Ks[128][KPADS];
  int bh = blockIdx.x, b = bh >> 3, h = bh & 7;
  int n0 = blockIdx.y * 128;
  int t = threadIdx.x;
  // async-stage Q panel (48x64) and K tile (128x64), 16B chunks
  const float* qb = qred + (size_t)bh * MPAD * D_;
  for (int e = t; e < MPAD * 16; e += 256) {
    int r = e >> 4, c4 = e & 15;
    async_copy_b128(qb + r * D_ + c4 * 4, &Qs[r][c4 * 4]);
  }
  const float* kb = kmat + (((size_t)b * L_ + n0) * H_ + h) * D_;
  for (int e = t; e < 128 * 16; e += 256) {
    int r = e >> 4, c4 = e & 15;
    async_copy_b128(kb + (size_t)r * H_ * D_ + c4 * 4, &Ks[r][c4 * 4]);
  }
  async_wait();
  __syncthreads();
  int wave = t >> 5, lane = t & 31;
  int m15 = lane & 15;
  int kb2 = (lane >> 4) << 1;                // K sub-offset: 0 (lanes 0-15) / 2 (16-31)
  int nloc = wave * 16;                      // each wave: one 16-column strip
  float* srow = scores + (size_t)bh * MPAD * L_;
  for (int mt = 0; mt < 3; ++mt) {
    v8f acc = {};
    for (int kk = 0; kk < D_; kk += 4) {
      v2f a, bf;
      a.x  = Qs[mt * 16 + m15][kk + kb2];
      a.y  = Qs[mt * 16 + m15][kk + kb2 + 1];
      bf.x = Ks[nloc + m15][kk + kb2];
      bf.y = Ks[nloc + m15][kk + kb2 + 1];
      acc = wmma4(a, bf, acc);
    }
    int rbase = mt * 16 + ((lane >> 4) << 3); // D rows: r (lanes 0-15) / r+8 (16-31)
#pragma unroll
    for (int r = 0; r < 8; ++r)
      srow[(size_t)(rbase + r) * L_ + n0 + nloc + m15] = acc[r] * 0.125f; // 1/sqrt(64)
  }
}

// ------------------------------------------------ 7) softmax over 2048 keys
__global__ void k_softmax(float* __restrict__ scores) {
  __shared__ float red[256];
  int bh = blockIdx.x, u = blockIdx.y, t = threadIdx.x;
  float* row = scores + ((size_t)bh * MPAD + u) * L_;
  float loc[8];
  float mx = -1e30f;
#pragma unroll
  for (int i = 0; i < 8; ++i) { loc[i] = row[t + i * 256]; mx = fmaxf(mx, loc[i]); }
  red[t] = mx; __syncthreads();
  for (int s = 128; s > 0; s >>= 1) { if (t < s) red[t] = fmaxf(red[t], red[t + s]); __syncthreads(); }
  mx = red[0]; __syncthreads();
  float sum = 0.f;
#pragma unroll
  for (int i = 0; i < 8; ++i) { loc[i] = __expf(loc[i] - mx); sum += loc[i]; }
  red[t] = sum; __syncthreads();
  for (int s = 128; s > 0; s >>= 1) { if (t < s) red[t] += red[t + s]; __syncthreads(); }
  float inv = 1.0f / red[0];
#pragma unroll
  for (int i = 0; i < 8; ++i) row[t + i * 256] = loc[i] * inv;
}

// -- 8) update = attn(48x2048) * V(2048x64), double-buffered async + WMMA,
//    then scatter the 40 selected rows into the output
__global__ void k_update(const float* __restrict__ vmat, const float* __restrict__ scores,
                         const int* __restrict__ mtop, float* __restrict__ out) {
  __shared__ __align__(16) float As[2][MPAD][KPADS];  // attn chunk (48 x 64)
  __shared__ __align__(16) float Vs[2][64][KPADS];    // V chunk (64 x 64)
  int bh = blockIdx.x, b = bh >> 3, h = bh & 7;
  int t = threadIdx.x, wave = t >> 5, lane = t & 31;
  int m15 = lane & 15, kb2 = (lane >> 4) << 1;
  // 12 output tiles (3 M x 4 N); wave w owns tiles {w, w+8}
  int tile0 = wave, tile1 = wave + 8;
  int mt0 = tile0 >> 2, nt0 = tile0 & 3;
  int mt1 = tile1 >> 2, nt1 = tile1 & 3;
  v8f acc0 = {}, acc1 = {};
  const float* sbase = scores + (size_t)bh * MPAD * L_;
  const float* vbase = vmat + ((size_t)b * L_ * H_ + h) * D_;

  // prologue: issue async loads for chunk 0
  for (int e = t; e < MPAD * 16; e += 256) {
    int r = e >> 4, c4 = e & 15;
    async_copy_b128(sbase + (size_t)r * L_ + c4 * 4, &As[0][r][c4 * 4]);
  }
  for (int e = t; e < 64 * 16; e += 256) {
    int r = e >> 4, c4 = e & 15;
    async_copy_b128(vbase + (size_t)r * H_ * D_ + c4 * 4, &Vs[0][r][c4 * 4]);
  }

  const int NCHUNK = L_ / 64;                // 32 contraction chunks
  for (int c = 0; c < NCHUNK; ++c) {
    int cur = c & 1, nxt = cur ^ 1;
    async_wait();                            // my wave's copies for chunk c done
    __syncthreads();                         // everyone's copies visible; nxt buffer free
    if (c + 1 < NCHUNK) {                    // overlap: issue chunk c+1 while computing c
      const float* sb = sbase + (c + 1) * 64;
      for (int e = t; e < MPAD * 16; e += 256) {
        int r = e >> 4, c4 = e & 15;
        async_copy_b128(sb + (size_t)r * L_ + c4 * 4, &As[nxt][r][c4 * 4]);
      }
      const float* vb = vbase + (size_t)(c + 1) * 64 * H_ * D_;
      for (int e = t; e < 64 * 16; e += 256) {
        int r = e >> 4, c4 = e & 15;
        async_copy_b128(vb + (size_t)r * H_ * D_ + c4 * 4, &Vs[nxt][r][c4 * 4]);
      }
    }
    for (int kk = 0; kk < 64; kk += 4) {
      v2f a, bf;
      a.x  = As[cur][mt0 * 16 + m15][kk + kb2];
      a.y  = As[cur][mt0 * 16 + m15][kk + kb2 + 1];
      bf.x = Vs[cur][kk + kb2][nt0 * 16 + m15];
      bf.y = Vs[cur][kk + kb2 + 1][nt0 * 16 + m15];
      acc0 = wmma4(a, bf, acc0);
    }
    if (tile1 < 12) {
      for (int kk = 0; kk < 64; kk += 4) {
        v2f a, bf;
        a.x  = As[cur][mt1 * 16 + m15][kk + kb2];
        a.y  = As[cur][mt1 * 16 + m15][kk + kb2 + 1];
        bf.x = Vs[cur][kk + kb2][nt1 * 16 + m15];
        bf.y = Vs[cur][kk + kb2 + 1][nt1 * 16 + m15];
        acc1 = wmma4(a, bf, acc1);
      }
    }
  }
  // scatter owned tiles into the selected output rows
#pragma unroll 1
  for (int which = 0; which < 2; ++which) {
    int tile = which ? tile1 : tile0;
    if (tile >= 12) continue;
    v8f acc = which ? acc1 : acc0;
    int mt = tile >> 2, nt = tile & 3;
    int rbase = mt * 16 + ((lane >> 4) << 3);
#pragma unroll
    for (int r = 0; r < 8; ++r) {
      int row = rbase + r;
      if (row < NTOP) {
        int l = mtop[bh * MPAD + row];
        out[(((size_t)b * L_ + l) * H_ + h) * D_ + nt * 16 + m15] = acc[r];
      }
    }
  }
}

// ------------------------------------------------------------------- launch
extern "C" void kernel_launch(void* const* d_in, const int* in_sizes, int n_in,
                              void* d_out, int out_size, void* d_ws, size_t ws_size,
                              hipStream_t stream) {
  const float* q = (const float*)d_in[0];
  const float* k = (const float*)d_in[1];
  const float* v = (const float*)d_in[2];
  float* out = (float*)d_out;
  float* ws  = (float*)d_ws;

  const int BH = B_ * H_;
  float* vmean  = ws;                                  // 32*64
  float* M      = vmean + BH * D_;                     // 32*2048
  int*   mtop   = (int*)(M + BH * L_);                 // 32*48 ints
  float* qred   = (float*)(mtop + BH * MPAD);          // 32*48*64
  float* scores = qred + (size_t)BH * MPAD * D_;       // 32*48*2048 (~12.6 MB total)

  k_vmean   <<<BH, 64, 0, stream>>>(v, vmean);
  k_broadcast<<<(B_ * L_ * H_ * D_) / 4 / 256, 256, 0, stream>>>((float4*)out, (const float4*)vmean);
  k_sample_m<<<dim3(BH, L_ / 8), 256, 0, stream>>>(q, k, M);
  k_topk    <<<BH, 256, 0, stream>>>(M, mtop);
  k_qred    <<<BH, 256, 0, stream>>>(q, mtop, qred);
  k_scores  <<<dim3(BH, L_ / 128), 256, 0, stream>>>(k, qred, scores);
  k_softmax <<<dim3(BH, NTOP), 256, 0, stream>>>(scores);
  k_update  <<<BH, 256, 0, stream>>>(v, scores, mtop, out);
}